// StandGCN2_15839839387790
// MI455X (gfx1250) — compile-verified
//
#include <hip/hip_runtime.h>
#include <hip/hip_bf16.h>

typedef float v2f __attribute__((ext_vector_type(2)));
typedef float v8f __attribute__((ext_vector_type(8)));

// ---------------------------------------------------------------------------
// Degree / normalization
// ---------------------------------------------------------------------------
__global__ void deg_init_kernel(float* __restrict__ deg, int n) {
    int i = blockIdx.x * blockDim.x + threadIdx.x;
    if (i < n) deg[i] = 1.0f;  // self-loop
}

__global__ void deg_accum_kernel(const long long* __restrict__ dst,
                                 float* __restrict__ deg, int e) {
    int i = blockIdx.x * blockDim.x + threadIdx.x;
    int stride = gridDim.x * blockDim.x;
    for (; i < e; i += stride) {
        atomicAdd(&deg[(int)dst[i]], 1.0f);
    }
}

__global__ void inv_sqrt_kernel(float* __restrict__ deg_inv, int n) {
    int i = blockIdx.x * blockDim.x + threadIdx.x;
    if (i < n) deg_inv[i] = rsqrtf(deg_inv[i]);  // in place: deg -> 1/sqrt(deg)
}

// Deterministic ReLU + dropout(p=0.5), scale 1/keep = 2.
__device__ __forceinline__ float relu_drop(float v, unsigned idx) {
    v = fmaxf(v, 0.0f);
    unsigned u = idx * 747796405u + 2891336453u;  // PCG-style hash
    u = ((u >> ((u >> 28u) + 4u)) ^ u) * 277803737u;
    u = (u >> 22u) ^ u;
    return (u & 1u) ? v * 2.0f : 0.0f;
}

// ---------------------------------------------------------------------------
// GEMM1: h0 = x @ W1   (Mx128 @ 128x64), fp32 WMMA 16x16x4.
// One wave computes a full 16x64 row-block: A fragment loaded once per
// k-step (8B vector load), 4 independent WMMAs against 4 B column tiles.
// B is staged in LDS in k-pair-interleaved layout so each B fragment is one
// naturally-paired ds_load_b64; row stride 160 (= 32 mod 64 words) makes the
// two half-waves cover all 64 LDS banks exactly once (conflict-free).
// Fused epilogue: agg = h0 * self_norm + b1   (self-loop term + bias)
// ---------------------------------------------------------------------------
#define S1 160  // LDS row stride (words) per k-pair, GEMM1
__global__ void gemm1_wmma_kernel(const float* __restrict__ x,    // [N,128]
                                  const float* __restrict__ W1,   // [128,64]
                                  const float* __restrict__ b1,   // [64]
                                  const float* __restrict__ invs, // [N]
                                  float* __restrict__ h0,         // [N,64]
                                  float* __restrict__ agg,        // [N,64]
                                  int n, int row_tiles) {
    __shared__ float sW[64 * S1];  // 64 k-pairs x 160 words = 40 KB

    const int tid = threadIdx.x;
    // Stage W1: sW[kp*S1 + c*2 + 0/1] = W1[(2kp+0/1)*64 + c]
    for (int i = tid; i < 64 * 64; i += 256) {
        const int kp = i >> 6, c = i & 63;
        v2f w;
        w.x = W1[(2 * kp) * 64 + c];
        w.y = W1[(2 * kp + 1) * 64 + c];
        *(v2f*)&sW[kp * S1 + c * 2] = w;
    }
    __syncthreads();

    const int wave = tid >> 5;
    const int lane = tid & 31;
    const int tile = blockIdx.x * 8 + wave;  // row tile index
    if (tile >= row_tiles) return;           // wave-uniform exit, EXEC all-1s

    const int tm = tile << 4;                 // row base
    const int laneHi = lane >> 4;             // 0: K pair {0,1}, 1: K pair {2,3}
    const int lane16 = lane & 15;
    const int arow = min(tm + lane16, n - 1); // clamp: branch-free inner loop

    v8f acc[4] = {v8f{}, v8f{}, v8f{}, v8f{}};
#pragma unroll 4
    for (int k0 = 0; k0 < 128; k0 += 4) {
        const int ka = k0 + laneHi * 2;
        const int kp = ka >> 1;  // k-pair row in LDS
        const v2f a = *(const v2f*)(x + arow * 128 + ka);  // 8B vector load
#pragma unroll
        for (int c = 0; c < 4; ++c) {
            const v2f b = *(const v2f*)&sW[kp * S1 + (c * 16 + lane16) * 2];
            acc[c] = __builtin_amdgcn_wmma_f32_16x16x4_f32(
                false, a, false, b, (short)0, acc[c], false, false);
        }
    }

#pragma unroll
    for (int r = 0; r < 8; ++r) {
        const int row = tm + r + 8 * laneHi;
        if (row < n) {
            float ss = invs[row];
            ss = ss * ss;  // self_norm
#pragma unroll
            for (int c = 0; c < 4; ++c) {
                const int col = c * 16 + lane16;
                const float v = acc[c][r];
                h0[row * 64 + col] = v;
                agg[row * 64 + col] = v * ss + b1[col];  // self-loop + bias
            }
        }
    }
}

// ---------------------------------------------------------------------------
// Edge scatter, layer 1: agg[dst] += h0[src] * inv[src]*inv[dst]  (64 feat)
// One wave per edge, lane f covers features f and f+32 (coalesced row).
// ---------------------------------------------------------------------------
__global__ void edge_scatter64_kernel(const long long* __restrict__ src,
                                      const long long* __restrict__ dst,
                                      const float* __restrict__ invs,
                                      const float* __restrict__ h0,
                                      float* __restrict__ agg, int e) {
    const int gid = blockIdx.x * blockDim.x + threadIdx.x;
    const int lane = gid & 31;
    const int wstride = (gridDim.x * blockDim.x) >> 5;
    for (int ed = gid >> 5; ed < e; ed += wstride) {
        const int s = (int)src[ed];
        const int d = (int)dst[ed];
        const int ne = ed + wstride;
        if (ne < e) {  // prefetch next source row -> global_prefetch_b8
            __builtin_prefetch(&h0[(int)src[ne] * 64 + lane], 0, 1);
        }
        const float nrm = invs[s] * invs[d];
        const float v0 = h0[s * 64 + lane];
        const float v1 = h0[s * 64 + lane + 32];
        atomicAdd(&agg[d * 64 + lane], v0 * nrm);
        atomicAdd(&agg[d * 64 + lane + 32], v1 * nrm);
    }
}

// ---------------------------------------------------------------------------
// GEMM2: h2 = relu_dropout(agg) @ W2  (Mx64 @ 64x40), fp32 WMMA 16x16x4.
// ReLU+dropout fused into the A-fragment load (each h element is consumed
// exactly once as an A fragment). One wave computes a 16x48 row-block
// (3 col tiles, cols >= 40 masked). B staged in LDS k-pair-interleaved,
// row stride 96 words (= 32 mod 64): conflict-free ds_load_b64 fragments.
// Fused epilogue: out = h2 * self_norm + b2  written straight to d_out.
// ---------------------------------------------------------------------------
#define S2 96  // LDS row stride (words) per k-pair, GEMM2
__global__ void gemm2_wmma_kernel(const float* __restrict__ hpre, // [N,64] pre-activation
                                  const float* __restrict__ W2,   // [64,40]
                                  const float* __restrict__ b2,   // [40]
                                  const float* __restrict__ invs, // [N]
                                  float* __restrict__ h2,         // [N,40]
                                  float* __restrict__ out,        // [N,40]
                                  int n, int row_tiles) {
    __shared__ float sW[32 * S2];  // 32 k-pairs x 96 words = 12 KB

    const int tid = threadIdx.x;
    // Stage W2 zero-padded to 48 cols: sW[kp*S2 + c*2 + 0/1]
    for (int i = tid; i < 32 * 48; i += 256) {
        const int kp = i / 48, c = i - kp * 48;
        v2f w;
        w.x = (c < 40) ? W2[(2 * kp) * 40 + c] : 0.0f;
        w.y = (c < 40) ? W2[(2 * kp + 1) * 40 + c] : 0.0f;
        *(v2f*)&sW[kp * S2 + c * 2] = w;
    }
    __syncthreads();

    const int wave = tid >> 5;
    const int lane = tid & 31;
    const int tile = blockIdx.x * 8 + wave;
    if (tile >= row_tiles) return;

    const int tm = tile << 4;
    const int laneHi = lane >> 4;
    const int lane16 = lane & 15;
    const int arow = min(tm + lane16, n - 1);

    v8f acc[3] = {v8f{}, v8f{}, v8f{}};
#pragma unroll 4
    for (int k0 = 0; k0 < 64; k0 += 4) {
        const int ka = k0 + laneHi * 2;
        const int kp = ka >> 1;
        const unsigned idx = (unsigned)(arow * 64 + ka);
        v2f a = *(const v2f*)(hpre + arow * 64 + ka);
        a.x = relu_drop(a.x, idx);      // fused activation + dropout mask
        a.y = relu_drop(a.y, idx + 1);
#pragma unroll
        for (int c = 0; c < 3; ++c) {
            const v2f b = *(const v2f*)&sW[kp * S2 + (c * 16 + lane16) * 2];
            acc[c] = __builtin_amdgcn_wmma_f32_16x16x4_f32(
                false, a, false, b, (short)0, acc[c], false, false);
        }
    }

#pragma unroll
    for (int r = 0; r < 8; ++r) {
        const int row = tm + r + 8 * laneHi;
        if (row < n) {
            float ss = invs[row];
            ss = ss * ss;
#pragma unroll
            for (int c = 0; c < 3; ++c) {
                const int col = c * 16 + lane16;
                if (col < 40) {
                    const float v = acc[c][r];
                    h2[row * 40 + col] = v;
                    out[row * 40 + col] = v * ss + b2[col];  // init with self term
                }
            }
        }
    }
}

// ---------------------------------------------------------------------------
// Edge scatter, layer 2: out[dst] += h2[src] * norm  (40 features)
// One wave per edge; lane covers f=lane, and f=lane+32 for lane<8.
// ---------------------------------------------------------------------------
__global__ void edge_scatter40_kernel(const long long* __restrict__ src,
                                      const long long* __restrict__ dst,
                                      const float* __restrict__ invs,
                                      const float* __restrict__ h2,
                                      float* __restrict__ out, int e) {
    const int gid = blockIdx.x * blockDim.x + threadIdx.x;
    const int lane = gid & 31;
    const int wstride = (gridDim.x * blockDim.x) >> 5;
    for (int ed = gid >> 5; ed < e; ed += wstride) {
        const int s = (int)src[ed];
        const int d = (int)dst[ed];
        const int ne = ed + wstride;
        if (ne < e) {
            __builtin_prefetch(&h2[(int)src[ne] * 40 + lane], 0, 1);
        }
        const float nrm = invs[s] * invs[d];
        const float v0 = h2[s * 40 + lane];
        atomicAdd(&out[d * 40 + lane], v0 * nrm);
        if (lane < 8) {
            const float v1 = h2[s * 40 + 32 + lane];
            atomicAdd(&out[d * 40 + 32 + lane], v1 * nrm);
        }
    }
}

// ---------------------------------------------------------------------------
// Launch
// ---------------------------------------------------------------------------
extern "C" void kernel_launch(void* const* d_in, const int* in_sizes, int n_in,
                              void* d_out, int out_size, void* d_ws, size_t ws_size,
                              hipStream_t stream) {
    const float* x        = (const float*)d_in[0];       // [N,128]
    const long long* ei   = (const long long*)d_in[1];   // [2,E] int64
    const float* W1       = (const float*)d_in[2];       // [128,64]
    const float* b1       = (const float*)d_in[3];       // [64]
    const float* W2       = (const float*)d_in[4];       // [64,40]
    const float* b2       = (const float*)d_in[5];       // [40]
    float* out            = (float*)d_out;               // [N,40]

    const int N = in_sizes[0] / 128;
    const int E = in_sizes[1] / 2;
    const long long* src = ei;
    const long long* dst = ei + E;

    // Workspace layout (256B aligned):
    //   invs [N] f32 | agg [N*64] f32 | h0/h2 [N*64] f32
    auto align256 = [](size_t v) { return (v + 255) & ~size_t(255); };
    char* ws = (char*)d_ws;
    size_t off = 0;
    float* invs = (float*)(ws + off);  off = align256(off + (size_t)N * 4);
    float* agg  = (float*)(ws + off);  off = align256(off + (size_t)N * 64 * 4);
    float* h0   = (float*)(ws + off);  off = align256(off + (size_t)N * 64 * 4);
    float* h2   = h0;  // reuse after h0 is consumed
    (void)ws_size; (void)n_in; (void)out_size;

    // 1) deg = 1 + scatter count over dst; invs = rsqrt(deg)
    deg_init_kernel<<<(N + 255) / 256, 256, 0, stream>>>(invs, N);
    deg_accum_kernel<<<(E + 255) / 256, 256, 0, stream>>>(dst, invs, E);
    inv_sqrt_kernel<<<(N + 255) / 256, 256, 0, stream>>>(invs, N);

    // 2) GEMM1 (WMMA fp32), wave = 16x64 row-block, fused self-loop/bias
    const int rowTiles = (N + 15) / 16;
    gemm1_wmma_kernel<<<(rowTiles + 7) / 8, 256, 0, stream>>>(
        x, W1, b1, invs, h0, agg, N, rowTiles);

    // 3) edge scatter layer 1 (one wave / edge)
    edge_scatter64_kernel<<<(E + 7) / 8, 256, 0, stream>>>(
        src, dst, invs, h0, agg, E);

    // 4) GEMM2 (WMMA fp32) with fused ReLU+dropout on A-load; writes h2 and
    //    initializes out = h2*self + b2
    gemm2_wmma_kernel<<<(rowTiles + 7) / 8, 256, 0, stream>>>(
        agg, W2, b2, invs, h2, out, N, rowTiles);

    // 5) edge scatter layer 2 into out
    edge_scatter40_kernel<<<(E + 7) / 8, 256, 0, stream>>>(
        src, dst, invs, h2, out, E);
}